// LNCC_61993557950680
// MI455X (gfx1250) — compile-verified
//
#include <hip/hip_runtime.h>
#include <hip/hip_bf16.h>
#include <math.h>

// ---------------------------------------------------------------------------
// Fused LNCC on MI455X (gfx1250, wave32).
//  * One wave per 16x16x4 output tile; M,R tile + halo staged in LDS via
//    global_load_async_to_lds_b32 (zero-fill first -> zero padding for free,
//    per-lane EXEC predication skips out-of-volume elements).
//  * W-axis 5-tap box pass on the matrix core: 16-line x 20-col field tile
//    times a 20x16 banded matrix (weights 1/5) as five accumulating
//    V_WMMA_F32_16X16X4_F32 ops. Field loop fully unrolled so the field
//    combine is straight-line code (no per-WMMA scalar branching).
//  * H/D passes are 5-sums in LDS (flat, shift/add addressing only);
//    final 1/25 scale restores the 1/125 box mean.
//  * corr pointwise, wave shfl_xor reduction, one f64 global atomic per wave.
// ---------------------------------------------------------------------------

typedef __attribute__((ext_vector_type(2))) float v2f;
typedef __attribute__((ext_vector_type(8))) float v8f;

#define AS(n) __attribute__((address_space(n)))

#define BATCH 2
#define DD    160
#define HH    192
#define WW    192
#define TD    4            // d-slices of output per tile
#define EPS_F 1e-5f

#define DCH (DD / TD)      // 40
#define HCH (HH / 16)      // 12
#define WCH (WW / 16)      // 12
#define NTILES (BATCH * DCH * HCH * WCH)   // 11520

#define NLINES ((TD + 4) * 20)             // 160 (z,h) lines per tile
#define STAGE  (NLINES * 20)               // 3200 floats per staged field

#if __has_builtin(__builtin_amdgcn_global_load_async_to_lds_b32) && \
    __has_builtin(__builtin_amdgcn_s_wait_asynccnt)
#define HAVE_ASYNC_LDS 1
#else
#define HAVE_ASYNC_LDS 0
#endif

__global__ void lncc_init(double* acc) { *acc = 0.0; }

__global__ __launch_bounds__(32)
void lncc_main(const float* __restrict__ M,
               const float* __restrict__ R,
               double* __restrict__ acc)
{
    // Flat LDS, line-major: all hot-loop addressing is shift/add.
    __shared__ float sMf[STAGE];               // [ (z*20+y)*20 + x ]
    __shared__ float sRf[STAGE];
    __shared__ float tWf[NLINES * 16];         // [ (z*20+y)*16 + n ]
    __shared__ float tHf[(TD + 4) * 256];      // [ z*256 + y*16 + x ]
    __shared__ float oFf[5 * TD * 256];        // [ f*1024 + z*256 + y*16 + x ]

    int t = blockIdx.x;
    const int wc = t % WCH; t /= WCH;
    const int hc = t % HCH; t /= HCH;
    const int dc = t % DCH; t /= DCH;
    const int b  = t;
    const int d0 = dc * TD, h0 = hc * 16, w0 = wc * 16;
    const int lane = threadIdx.x;

    // ---- stage M,R tile + 2-wide halo into LDS (zero padded) ----
#if HAVE_ASYNC_LDS
    for (int i = lane; i < STAGE; i += 32) { sMf[i] = 0.f; sRf[i] = 0.f; }
    __syncthreads();   // zero-fill visible before async writes land on top
    for (int i = lane; i < STAGE; i += 32) {
        int z = i / 400, rem = i % 400;
        int y = rem / 20, x = rem % 20;
        int d = d0 + z - 2, h = h0 + y - 2, w = w0 + x - 2;
        if (d >= 0 && d < DD && h >= 0 && h < HH && w >= 0 && w < WW) {
            size_t idx = (((size_t)b * DD + d) * HH + h) * WW + w;
            __builtin_amdgcn_global_load_async_to_lds_b32(
                (AS(1) int*)(M + idx), (AS(3) int*)(sMf + i), 0, 0);
            __builtin_amdgcn_global_load_async_to_lds_b32(
                (AS(1) int*)(R + idx), (AS(3) int*)(sRf + i), 0, 0);
        }
    }
    __builtin_amdgcn_s_wait_asynccnt(0);
    __syncthreads();
#else
    for (int i = lane; i < STAGE; i += 32) {
        int z = i / 400, rem = i % 400;
        int y = rem / 20, x = rem % 20;
        int d = d0 + z - 2, h = h0 + y - 2, w = w0 + x - 2;
        float mv = 0.f, rv = 0.f;
        if (d >= 0 && d < DD && h >= 0 && h < HH && w >= 0 && w < WW) {
            size_t idx = (((size_t)b * DD + d) * HH + h) * WW + w;
            mv = M[idx];
            rv = R[idx];
        }
        sMf[i] = mv;
        sRf[i] = rv;
    }
    __syncthreads();
#endif

    // A-matrix (16x4 f32) per-lane layout: M-row = lane&15,
    // K pair = {0,1} for lanes 0-15, {2,3} for lanes 16-31.
    const int mrow = lane & 15;
    const int kb   = (lane >> 4) << 1;
    const int hi   = (lane >> 4) * 8;   // D-matrix M offset for this half-wave

    // Banded 20x16 B-matrix (weights 1/5), hoisted: depends only on kc/kb/mrow.
    // S[j][n] = 0.2 iff n <= j <= n+4
    v2f band[5];
    #pragma unroll
    for (int kc = 0; kc < 5; ++kc) {
        int k0 = kc * 4 + kb;
        float b0 = (k0     >= mrow && k0     <= mrow + 4) ? 0.2f : 0.f;
        float b1 = (k0 + 1 >= mrow && k0 + 1 <= mrow + 4) ? 0.2f : 0.f;
        band[kc] = (v2f){b0, b1};
    }

    #pragma unroll
    for (int f = 0; f < 5; ++f) {
        // ---- W pass: banded-matrix conv on the matrix core ----
        // 160 (z,h) lines -> 10 groups of 16; each group: 5x WMMA f32 16x16x4.
        for (int g = 0; g < NLINES / 16; ++g) {
            const int aBase = (g * 16 + mrow) * 20;   // this lane's A row
            v8f cacc = {};
            #pragma unroll
            for (int kc = 0; kc < 5; ++kc) {
                int k0 = kc * 4 + kb;
                float a0, a1;
                if (f == 0) {
                    a0 = sMf[aBase + k0];
                    a1 = sMf[aBase + k0 + 1];
                } else if (f == 1) {
                    a0 = sRf[aBase + k0];
                    a1 = sRf[aBase + k0 + 1];
                } else if (f == 2) {
                    float m0 = sMf[aBase + k0], m1 = sMf[aBase + k0 + 1];
                    a0 = m0 * m0; a1 = m1 * m1;
                } else if (f == 3) {
                    float r0 = sRf[aBase + k0], r1 = sRf[aBase + k0 + 1];
                    a0 = r0 * r0; a1 = r1 * r1;
                } else {
                    float m0 = sMf[aBase + k0], m1 = sMf[aBase + k0 + 1];
                    float r0 = sRf[aBase + k0], r1 = sRf[aBase + k0 + 1];
                    a0 = m0 * r0; a1 = m1 * r1;
                }
                v2f A = {a0, a1};
                cacc = __builtin_amdgcn_wmma_f32_16x16x4_f32(
                    /*neg_a=*/false, A, /*neg_b=*/false, band[kc],
                    /*c_mod=*/(short)0, cacc,
                    /*reuse_a=*/false, /*reuse_b=*/false);
            }
            // D layout: VGPR r holds M-row = r + hi, N = mrow.
            const int sBase = (g * 16 + hi) * 16 + mrow;
            #pragma unroll
            for (int r8 = 0; r8 < 8; ++r8)
                tWf[sBase + (r8 << 4)] = cacc[r8];
        }
        __syncthreads();

        // ---- H pass: 5-sum over rows (lines z*20+y .. z*20+y+4) ----
        for (int i = lane; i < (TD + 4) * 256; i += 32) {
            int lb = (i >> 8) * 320 + (i & 255);   // (z*20+y)*16 + x
            tHf[i] = tWf[lb]      + tWf[lb + 16] + tWf[lb + 32]
                   + tWf[lb + 48] + tWf[lb + 64];
        }
        __syncthreads();

        // ---- D pass: 5-sum over slices, scale 1/25 (W pass carried 1/5) ----
        for (int i = lane; i < TD * 256; i += 32) {
            oFf[f * 1024 + i] = 0.04f * (tHf[i]       + tHf[i + 256]
                                       + tHf[i + 512] + tHf[i + 768]
                                       + tHf[i + 1024]);
        }
        __syncthreads();
    }

    // ---- pointwise corr + wave reduction ----
    float part = 0.f;
    for (int i = lane; i < TD * 256; i += 32) {
        float Mm  = oFf[i];
        float Rm  = oFf[1024 + i];
        float MMm = oFf[2048 + i];
        float RRm = oFf[3072 + i];
        float MRm = oFf[4096 + i];
        float Mv = sqrtf(fmaxf(MMm - Mm * Mm, 0.f) + EPS_F);
        float Rv = sqrtf(fmaxf(RRm - Rm * Rm, 0.f) + EPS_F);
        part += (MRm - Mm * Rm) / (Mv * Rv + EPS_F);
    }
    for (int off = 16; off > 0; off >>= 1)
        part += __shfl_xor(part, off, 32);
    if (lane == 0)
        atomicAdd(acc, (double)part);
}

__global__ void lncc_fin(const double* __restrict__ acc, float* __restrict__ out)
{
    const double N = (double)BATCH * DD * HH * WW;
    out[0] = (float)(-(*acc) / N);
}

extern "C" void kernel_launch(void* const* d_in, const int* in_sizes, int n_in,
                              void* d_out, int out_size, void* d_ws, size_t ws_size,
                              hipStream_t stream)
{
    const float* M = (const float*)d_in[0];
    const float* R = (const float*)d_in[1];
    float* out = (float*)d_out;
    double* acc = (double*)d_ws;   // 8-byte f64 accumulator in scratch

    lncc_init<<<1, 1, 0, stream>>>(acc);
    lncc_main<<<NTILES, 32, 0, stream>>>(M, R, acc);
    lncc_fin<<<1, 1, 0, stream>>>(acc, out);

    (void)in_sizes; (void)n_in; (void)out_size; (void)ws_size;
}